// Radon_644245094841
// MI455X (gfx1250) — compile-verified
//
#include <hip/hip_runtime.h>
#include <math.h>

typedef __attribute__((ext_vector_type(2))) float v2f;
typedef __attribute__((ext_vector_type(8))) float v8f;

#define IMG_W 512
#define IMG_H 512
#define N_ANG 180

// One bilinear sample with the reference's border semantics:
// clamp the index, zero the weight if the (unclamped) tap is outside.
__device__ __forceinline__ float bilin_sample(const float* __restrict__ img,
                                              float iy, float ix) {
  float x0 = floorf(ix);
  float y0 = floorf(iy);
  float wx1 = ix - x0, wy1 = iy - y0;
  float wx0 = 1.f - wx1, wy0 = 1.f - wy1;
  float acc = 0.f;
#pragma unroll
  for (int dy = 0; dy < 2; ++dy) {
#pragma unroll
    for (int dx = 0; dx < 2; ++dx) {
      float xf = x0 + (float)dx;
      float yf = y0 + (float)dy;
      float w  = (dy ? wy1 : wy0) * (dx ? wx1 : wx0);
      bool valid = (xf >= 0.f) && (xf <= (float)(IMG_W - 1)) &&
                   (yf >= 0.f) && (yf <= (float)(IMG_H - 1));
      int ic = (int)fminf(fmaxf(xf, 0.f), (float)(IMG_W - 1));
      int jc = (int)fminf(fmaxf(yf, 0.f), (float)(IMG_H - 1));
      acc = fmaf(img[jc * IMG_W + ic], valid ? w : 0.f, acc);
    }
  }
  return acc;
}

// One wave handles (batch nb, angle a, 16 detector bins). The j-loop walks the
// 512 samples along the ray; every 4 j's are reduced on the matrix pipe via
// V_WMMA_F32_16X16X4_F32 with an all-ones A (D[m,n] = sum_k B[k,n] + C[m,n]).
__global__ void radon_wmma_kernel(const float* __restrict__ x,
                                  const float* __restrict__ theta,
                                  float* __restrict__ out) {
  const int lane = threadIdx.x & 31;
  const int wave = threadIdx.x >> 5;
  const int a  = blockIdx.y;          // angle
  const int nb = blockIdx.z;          // batch
  const int col = lane & 15;          // N index within 16x16 tile
  const int grp = lane >> 4;          // lane half -> which K pair we supply
  const int i  = blockIdx.x * 128 + wave * 16 + col;  // detector bin

  const float rad = theta[a] * 0.017453292519943295f; // pi/180
  const float c = cosf(rad);
  const float s = sinf(rad);

  // Xi = -1 + i*2/511 ; ix = 255.5*(c*Xi + s*Yj + 1), Yj = -1 + j*2/511
  //  => ix(j) = 255.5*(c*Xi - s + 1) + j*s
  //     iy(j) = 255.5*(-s*Xi - c + 1) + j*c
  const float Xi  = fmaf((float)i, 2.f / 511.f, -1.f);
  const float ixb = 255.5f * fmaf(c, Xi, 1.f - s);
  const float iyb = 255.5f * fmaf(-s, Xi, 1.f - c);

  const float* __restrict__ img = x + (size_t)nb * (IMG_H * IMG_W);

  v2f ones;
  ones.x = 1.0f;
  ones.y = 1.0f;
  v8f acc = {};

  // B layout for 32-bit 4x16: VGPR0 holds K=0 (lanes 0-15) / K=2 (lanes 16-31),
  // VGPR1 holds K=1 / K=3, N = lane%16. So this lane supplies rows
  // j0 + 2*grp and j0 + 2*grp + 1 of the 4-row tile.
  for (int j0 = 0; j0 < IMG_H; j0 += 4) {
    const float fja = (float)(j0 + 2 * grp);
    v2f b;
    b.x = bilin_sample(img, fmaf(fja,       c, iyb), fmaf(fja,       s, ixb));
    b.y = bilin_sample(img, fmaf(fja + 1.f, c, iyb), fmaf(fja + 1.f, s, ixb));
    acc = __builtin_amdgcn_wmma_f32_16x16x4_f32(
        /*neg_a=*/false, ones, /*neg_b=*/false, b,
        /*c_mod=*/(short)0, acc, /*reuse_a=*/false, /*reuse_b=*/false);
  }

  // Every row of D holds the same column sums; row M=0 lives in acc[0] on
  // lanes 0..15 with N = lane. Output layout: out[nb, 0, i, a].
  if (lane < 16) {
    out[((size_t)nb * IMG_W + (size_t)i) * N_ANG + a] = acc[0];
  }
}

extern "C" void kernel_launch(void* const* d_in, const int* in_sizes, int n_in,
                              void* d_out, int out_size, void* d_ws, size_t ws_size,
                              hipStream_t stream) {
  (void)in_sizes; (void)n_in; (void)out_size; (void)d_ws; (void)ws_size;
  const float* x     = (const float*)d_in[0];
  const float* theta = (const float*)d_in[1];
  float* out = (float*)d_out;

  dim3 block(256);                 // 8 waves; uniform control flow (EXEC all-1s)
  dim3 grid(IMG_W / 128, N_ANG, 2);
  radon_wmma_kernel<<<grid, block, 0, stream>>>(x, theta, out);
}